// ResidualBackwardQuantizer_79413945303138
// MI455X (gfx1250) — compile-verified
//
#include <hip/hip_runtime.h>

typedef float v2f __attribute__((ext_vector_type(2)));
typedef float v8f __attribute__((ext_vector_type(8)));

constexpr int N_ = 8, D_ = 8, H_ = 32, W_ = 32, K_ = 4096;
constexpr int HW = H_ * W_;          // 1024
constexpr int P_ = N_ * HW;          // 8192 pixels (m == 1)
constexpr float EPSf = 1e-6f;

// monotone float -> uint mapping (total order preserved)
__device__ __forceinline__ unsigned int mapOrd(float x) {
    unsigned int u = __float_as_uint(x);
    return (u & 0x80000000u) ? ~u : (u | 0x80000000u);
}
// pack (value, index) so that u64-max == (max value, min index on ties)
__device__ __forceinline__ unsigned long long packKey(float x, int j) {
    return ((unsigned long long)mapOrd(x) << 32) | (unsigned int)(~j);
}

__global__ void vq_init(unsigned long long* __restrict__ keys) {
    int i = blockIdx.x * blockDim.x + threadIdx.x;
    if (i < 2 * P_) keys[i] = 0ull;
}

__global__ __launch_bounds__(256) void vq_prep(const float* __restrict__ cb,
                                               const float* __restrict__ temp,
                                               const float* __restrict__ freq,
                                               float* __restrict__ c2,
                                               float* __restrict__ logF,
                                               float* __restrict__ scal) {
    __shared__ float red[256];
    int tid = threadIdx.x;
    float cnt = 0.f;
    for (int j = tid; j < K_; j += 256) {
        float f = freq[j];
        cnt += (f > EPSf) ? 1.f : 0.f;
        logF[j] = __logf(f);
        float s = 0.f;
#pragma unroll
        for (int dd = 0; dd < D_; ++dd) { float v = cb[j * D_ + dd]; s += v * v; }
        c2[j] = s;
    }
    red[tid] = cnt;
    __syncthreads();
    for (int off = 128; off > 0; off >>= 1) {
        if (tid < off) red[tid] += red[tid + off];
        __syncthreads();
    }
    if (tid == 0) {
        float cu = red[0] / (float)K_;
        cu = fminf(fmaxf(cu, 0.f), 1.f);
        const float bits = 12.0f;                       // log2(4096)
        scal[0] = -(bits - 1.0f) * cu * cu + bits;      // expo
        scal[1] = -fmaxf(temp[0], EPSf) * (1.0f / 64.0f); // -temp / sqrt(k)
    }
}

__global__ __launch_bounds__(256) void vq_main(const float* __restrict__ x,
                                               const float* __restrict__ cb,
                                               const float* __restrict__ dropu,
                                               const float* __restrict__ gumu,
                                               const float* __restrict__ c2,
                                               const float* __restrict__ logF,
                                               const float* __restrict__ scal,
                                               float* __restrict__ logitOut,
                                               unsigned long long* __restrict__ keyL,
                                               unsigned long long* __restrict__ keyG) {
    __shared__ float s_x2[16];
    const int tid    = threadIdx.x;
    const int lane   = tid & 31;
    const int waveId = tid >> 5;
    const int laneN  = lane & 15;   // code within tile / pixel row for A
    const int laneHi = lane >> 4;
    const int pixBase = blockIdx.x * 16;

    // ||x||^2 for this block's 16 pixels
    if (tid < 16) {
        int p  = pixBase + tid;
        int xb = (p >> 10) * (D_ * HW) + (p & (HW - 1));
        float s = 0.f;
#pragma unroll
        for (int kk = 0; kk < D_; ++kk) { float v = x[xb + kk * HW]; s += v * v; }
        s_x2[tid] = s;
    }
    __syncthreads();

    const float expo         = scal[0];
    const float negTempScale = scal[1];

    // A operand (16 pixels x K=8, split as two 16x4 f32 operands).
    // Layout: lanes 0-15 hold M=lane {K=0(V0),1(V1)}, lanes 16-31 hold {K=2,3}.
    const int pA = pixBase + laneN;
    const int xb = (pA >> 10) * (D_ * HW) + (pA & (HW - 1));
    const int ka = laneHi * 2;
    v2f aLo, aHi;
    aLo.x = x[xb + ka * HW];       aLo.y = x[xb + (ka + 1) * HW];
    aHi.x = x[xb + (ka + 4) * HW]; aHi.y = x[xb + (ka + 5) * HW];

    unsigned long long mk0[8], mk1[8];
#pragma unroll
    for (int v = 0; v < 8; ++v) { mk0[v] = 0ull; mk1[v] = 0ull; }

    for (int t = 0; t < K_ / 128; ++t) {
        const int j = t * 128 + waveId * 16 + laneN;
        // B operand (K=8 x 16 codes): codebook row j is 8 contiguous floats.
        const float* cbr = cb + j * D_ + ka;
        v2f bLo = *(const v2f*)(cbr);
        v2f bHi = *(const v2f*)(cbr + 4);

        v8f c = {};
        c = __builtin_amdgcn_wmma_f32_16x16x4_f32(false, aLo, false, bLo,
                                                  (short)0, c, false, false);
        c = __builtin_amdgcn_wmma_f32_16x16x4_f32(false, aHi, false, bHi,
                                                  (short)0, c, false, false);

        const float c2j = c2[j];
        const float lFj = logF[j];
#pragma unroll
        for (int v = 0; v < 8; ++v) {
            const int r = v + (laneHi << 3);                 // C layout: M = v + 8*(lane>=16)
            const size_t e = (size_t)(pixBase + r) * K_ + j;
            float du = __builtin_nontemporal_load(dropu + e);
            float gu = __builtin_nontemporal_load(gumu + e);
            float dist  = s_x2[r] + c2j - 2.0f * c[v];
            float logit = dist * negTempScale;
            // drop_u^expo < F  <=>  expo*log(u) < log(F)
            if (expo * __logf(du) < lFj) logit -= 1e9f;
            __builtin_nontemporal_store(logit, logitOut + e);
            float g = -__logf(-__logf(gu + 1e-20f) + 1e-20f);
            unsigned long long k0 = packKey(logit, j);
            unsigned long long k1 = packKey(logit + g, j);
            mk0[v] = mk0[v] > k0 ? mk0[v] : k0;
            mk1[v] = mk1[v] > k1 ? mk1[v] : k1;
        }
    }

#pragma unroll
    for (int v = 0; v < 8; ++v) {
        const int r = v + (laneHi << 3);
        atomicMax(&keyL[pixBase + r], mk0[v]);
        atomicMax(&keyG[pixBase + r], mk1[v]);
    }
}

__global__ void vq_final(const unsigned long long* __restrict__ keyL,
                         const unsigned long long* __restrict__ keyG,
                         const float* __restrict__ cb,
                         float* __restrict__ out) {
    int p = blockIdx.x * blockDim.x + threadIdx.x;
    if (p >= P_) return;
    int codeIdx = (int)((~(unsigned int)keyL[p]) & (K_ - 1));
    ((int*)(out + N_ * D_ * HW))[p] = codeIdx;          // code output (int32)
    int jq = (int)((~(unsigned int)keyG[p]) & (K_ - 1));
    int nIdx = p >> 10, s = p & (HW - 1);
    float* q = out + nIdx * (D_ * HW) + s;              // quantized [n, d, h, w]
    const float* row = cb + jq * D_;
#pragma unroll
    for (int dd = 0; dd < D_; ++dd) q[dd * HW] = row[dd];
}

extern "C" void kernel_launch(void* const* d_in, const int* in_sizes, int n_in,
                              void* d_out, int out_size, void* d_ws, size_t ws_size,
                              hipStream_t stream) {
    (void)in_sizes; (void)n_in; (void)out_size; (void)ws_size;
    const float* x    = (const float*)d_in[0];
    const float* cb   = (const float*)d_in[1];
    const float* temp = (const float*)d_in[2];
    const float* freq = (const float*)d_in[3];
    const float* du   = (const float*)d_in[4];
    const float* gu   = (const float*)d_in[5];
    float* out = (float*)d_out;

    unsigned long long* keyL = (unsigned long long*)d_ws;
    unsigned long long* keyG = keyL + P_;
    float* c2   = (float*)(keyG + P_);
    float* logF = c2 + K_;
    float* scal = logF + K_;

    float* logitOut = out + (N_ * D_ * HW) + P_;   // after quantized(65536) + code(8192)

    vq_init<<<(2 * P_ + 255) / 256, 256, 0, stream>>>(keyL);
    vq_prep<<<1, 256, 0, stream>>>(cb, temp, freq, c2, logF, scal);
    vq_main<<<P_ / 16, 256, 0, stream>>>(x, cb, du, gu, c2, logF, scal,
                                         logitOut, keyL, keyG);
    vq_final<<<(P_ + 255) / 256, 256, 0, stream>>>(keyL, keyG, cb, out);
}